// CTCLayer_33784212750820
// MI455X (gfx1250) — compile-verified
//
#include <hip/hip_runtime.h>

// CTC batch cost (tf.keras ctc_batch_cost, input_length=T, label_length=L).
// B=256, T=512, C=512 (blank=511), L=128, S=2L+1=257.
// Strategy: one workgroup per batch row; per-state thread; depth-8 async
// global->LDS prefetch of probability rows (ASYNCcnt pipeline); log2-domain
// forward recursion using raw v_exp_f32/v_log_f32; ln2 scale at the end.
#define CTC_B   256
#define CTC_T   512
#define CTC_C   512
#define CTC_L   128
#define CTC_S   257
#define BLOCK   288          // 9 wave32's; threads 0..256 own states
#define DEPTH   8            // async prefetch ring depth
#define BLANK   (CTC_C - 1)
#define EPSF    1e-7f
#define NEGF    (-1e30f)     // log-space -inf surrogate (domain-agnostic)
#define LN2F    0.6931471805599453f

// CDNA5 async global->LDS copy (ASYNCcnt-tracked), per CDNA5 ISA.
__device__ __forceinline__ void async_copy_b128(unsigned lds_off, const void* gaddr) {
    unsigned long long ga = (unsigned long long)gaddr;
    asm volatile("global_load_async_to_lds_b128 %0, %1, off"
                 :: "v"(lds_off), "v"(ga)
                 : "memory");
}
__device__ __forceinline__ void wait_async_le6() {
    asm volatile("s_wait_asynccnt 6" ::: "memory");   // DEPTH-2 outstanding ok
}

__global__ __launch_bounds__(BLOCK)
void ctc_alpha_kernel(const int* __restrict__ labels,   // [B, L] int32
                      const float* __restrict__ ypred,  // [B, T, C] probs
                      float* __restrict__ out) {        // [B] loss
    __shared__ float rowbuf[DEPTH][CTC_C];   // 8 x 2KB prob-row ring
    __shared__ float alpha[2][CTC_S + 2];    // +2 low pad = NEG
    __shared__ int   ext[CTC_S];

    const int b   = blockIdx.x;
    const int tid = threadIdx.x;
    const float* yp = ypred + (size_t)b * CTC_T * CTC_C;

    // Extended labels: blank, l0, blank, l1, ..., blank
    if (tid < CTC_S)
        ext[tid] = (tid & 1) ? labels[b * CTC_L + (tid >> 1)] : BLANK;
    if (tid >= CTC_S && tid < CTC_S + 4) {
        int k = tid - CTC_S;                 // 0..3 -> alpha[0..1][0..1] pads
        alpha[k >> 1][k & 1] = NEGF;
    }
    // Prefetch rows 0..DEPTH-2 (7 issues per loading wave)
    if (tid < 128) {
        #pragma unroll
        for (int r = 0; r < DEPTH - 1; ++r)
            async_copy_b128((unsigned)(size_t)&rowbuf[r][tid * 4],
                            yp + (size_t)r * CTC_C + tid * 4);
    }
    wait_async_le6();        // 7 issued, <=6 outstanding => row 0 resident
    __syncthreads();

    const int  e  = (tid < CTC_S) ? ext[tid] : 0;
    const bool sk = (tid < CTC_S) && (tid >= 3) && (tid & 1) && (ext[tid] != ext[tid - 2]);

    // Top off the ring with row DEPTH-1, then t=0 init (log2 domain).
    if (tid < 128)
        async_copy_b128((unsigned)(size_t)&rowbuf[DEPTH - 1][tid * 4],
                        yp + (size_t)(DEPTH - 1) * CTC_C + tid * 4);
    float a = NEGF;
    if (tid < CTC_S) {
        float lp0 = __builtin_amdgcn_logf(rowbuf[0][e] + EPSF);  // v_log_f32 (log2)
        a = (tid <= 1) ? lp0 : NEGF;
        alpha[0][tid + 2] = a;
    }

    for (int t = 1; t < CTC_T; ++t) {
        const int cur = t & 1;
        const int prv = cur ^ 1;

        wait_async_le6();            // row t resident (issued 7 iters ago)
        __syncthreads();             // row t + alpha[prv] visible; row t-1 reads done

        if (t + DEPTH - 1 < CTC_T && tid < 128)   // reuse slot of row t-1
            async_copy_b128((unsigned)(size_t)&rowbuf[(t + DEPTH - 1) & (DEPTH - 1)][tid * 4],
                            yp + (size_t)(t + DEPTH - 1) * CTC_C + tid * 4);

        if (tid < CTC_S) {
            float a1 = a;                            // own alpha in VGPR
            float a2 = alpha[prv][tid + 1];          // state s-1
            float a3 = sk ? alpha[prv][tid] : NEGF;  // state s-2
            float p  = rowbuf[t & (DEPTH - 1)][e] + EPSF;
            float m  = fmaxf(a1, fmaxf(a2, a3));
            float sm = __builtin_amdgcn_exp2f(a1 - m)    // v_exp_f32 x3
                     + __builtin_amdgcn_exp2f(a2 - m)
                     + __builtin_amdgcn_exp2f(a3 - m);
            // fold emission into the normalizer: log2(sm) + log2(p) = log2(sm*p)
            a = m + __builtin_amdgcn_logf(sm * p);       // v_log_f32 x1
            alpha[cur][tid + 2] = a;
        }
    }
    __syncthreads();

    if (tid == 0) {
        const int fin = (CTC_T - 1) & 1;
        float x = alpha[fin][CTC_S - 1 + 2];
        float y = alpha[fin][CTC_S - 2 + 2];
        float m = fmaxf(x, y);
        float l2 = m + __builtin_amdgcn_logf(__builtin_amdgcn_exp2f(x - m)
                                           + __builtin_amdgcn_exp2f(y - m));
        out[b] = -LN2F * l2;         // back to natural log
    }
}

extern "C" void kernel_launch(void* const* d_in, const int* in_sizes, int n_in,
                              void* d_out, int out_size, void* d_ws, size_t ws_size,
                              hipStream_t stream) {
    const int*   labels = (const int*)d_in[0];   // y_true [256,128] int32
    const float* ypred  = (const float*)d_in[1]; // y_pred [256,512,512] fp32
    float*       out    = (float*)d_out;         // [256,1] fp32
    (void)in_sizes; (void)n_in; (void)out_size; (void)d_ws; (void)ws_size;
    hipLaunchKernelGGL(ctc_alpha_kernel, dim3(CTC_B), dim3(BLOCK), 0, stream,
                       labels, ypred, out);
}